// MoNCELoss_15934328668294
// MI455X (gfx1250) — compile-verified
//
#include <hip/hip_runtime.h>
#include <math.h>

#define NP   2048
#define DIM  256
#define NB   8
#define NROW 16384
#define INV_T (1.0f/0.07f)
#define AW   (1.0f/2048.0f)

typedef float v2f __attribute__((ext_vector_type(2)));
typedef float v8f __attribute__((ext_vector_type(8)));

// ---------------- row norms of q and k (clamped like F.normalize) -------------
__global__ void norms_kernel(const float* __restrict__ q, const float* __restrict__ k,
                             float* __restrict__ rq, float* __restrict__ rk) {
    int row  = blockIdx.x * 8 + (threadIdx.x >> 5);   // 0..16383
    int lane = threadIdx.x & 31;
    const float* src = (blockIdx.y == 0) ? q : k;
    const float* p = src + (size_t)row * DIM;
    float s = 0.f;
#pragma unroll
    for (int i = 0; i < DIM / 32; ++i) { float x = p[lane + i * 32]; s += x * x; }
#pragma unroll
    for (int off = 16; off; off >>= 1) s += __shfl_xor(s, off, 32);
    if (lane == 0) {
        float n = fmaxf(sqrtf(s), 1e-12f);
        ((blockIdx.y == 0) ? rq : rk)[row] = n;
    }
}

// ---------------- init v = 1/n ------------------------------------------------
__global__ void init_v_kernel(float* __restrict__ v) {
    int i = blockIdx.x * blockDim.x + threadIdx.x;
    if (i < NROW) v[i] = AW;
}

// ---------------- S = q k^T via f32 WMMA, then K = exp(-S/(|q||k|)) -----------
// block: 256 threads = 8 waves; block tile 64(M) x 128(N); wave tile 32x32.
__global__ void gemm_expK_kernel(const float* __restrict__ q, const float* __restrict__ k,
                                 const float* __restrict__ rq, const float* __restrict__ rk,
                                 float* __restrict__ Kbuf) {
    int b     = blockIdx.z;
    int n_blk = blockIdx.x * 128;
    int m_blk = blockIdx.y * 64;
    int lane  = threadIdx.x & 31;
    int w     = threadIdx.x >> 5;
    int wm    = (w & 1) * 32;
    int wn    = (w >> 1) * 32;

    const float* qb = q + (size_t)b * NP * DIM;
    const float* kb = k + (size_t)b * NP * DIM;

    int l16   = lane & 15;
    int khalf = (lane >> 4) << 1;              // 0 (lanes 0-15) or 2 (lanes 16-31)
    int mrow0 = m_blk + wm + l16;              // A fragment rows (subtile 0 / +16)
    int ncol0 = n_blk + wn + l16;              // B fragment cols

    v8f acc00 = {}, acc01 = {}, acc10 = {}, acc11 = {};

    for (int kk = 0; kk < DIM; kk += 4) {
        // A: 16x4 f32, lane-half picks K pair, two dwords per lane (K, K+1)
        v2f a0 = *(const v2f*)(qb + (size_t)mrow0 * DIM + kk + khalf);
        v2f a1 = *(const v2f*)(qb + (size_t)(mrow0 + 16) * DIM + kk + khalf);
        // B: 4x16 f32, B[kd][n] = kfeat[n][kd] -> same per-lane pattern over k rows
        v2f b0 = *(const v2f*)(kb + (size_t)ncol0 * DIM + kk + khalf);
        v2f b1 = *(const v2f*)(kb + (size_t)(ncol0 + 16) * DIM + kk + khalf);

        acc00 = __builtin_amdgcn_wmma_f32_16x16x4_f32(false, a0, false, b0, (short)0, acc00, false, false);
        acc01 = __builtin_amdgcn_wmma_f32_16x16x4_f32(false, a0, false, b1, (short)0, acc01, false, false);
        acc10 = __builtin_amdgcn_wmma_f32_16x16x4_f32(false, a1, false, b0, (short)0, acc10, false, false);
        acc11 = __builtin_amdgcn_wmma_f32_16x16x4_f32(false, a1, false, b1, (short)0, acc11, false, false);
    }

    // D layout: lanes 0-15 -> M = elem, lanes 16-31 -> M = elem+8; N = lane&15
    int hi = (lane >> 4) * 8;
    float* Kb = Kbuf + (size_t)b * NP * NP;
    v8f accs[2][2] = { { acc00, acc01 }, { acc10, acc11 } };
#pragma unroll
    for (int ms = 0; ms < 2; ++ms) {
#pragma unroll
        for (int ns = 0; ns < 2; ++ns) {
            int ncol = n_blk + wn + ns * 16 + l16;
            float rkn = rk[b * NP + ncol];
#pragma unroll
            for (int i = 0; i < 8; ++i) {
                int mrow = m_blk + wm + ms * 16 + hi + i;
                float rqm = rq[b * NP + mrow];
                float C   = accs[ms][ns][i] / (rqm * rkn);   // cosine cost
                Kb[(size_t)mrow * NP + ncol] = __expf(-C);   // eps = 1
            }
        }
    }
}

// ---------------- Sinkhorn: u = a / (K v + 1e-8), wave per row ---------------
__global__ void sinkhorn_u_kernel(const float* __restrict__ Kbuf, const float* __restrict__ v,
                                  float* __restrict__ u) {
    int row  = blockIdx.x * 8 + (threadIdx.x >> 5);   // global row 0..16383
    int lane = threadIdx.x & 31;
    int b = row >> 11, m = row & (NP - 1);
    const float* Kb = Kbuf + (size_t)b * NP * NP + (size_t)m * NP;
    const float* vb = v + b * NP;
    float s = 0.f;
    for (int n = lane; n < NP; n += 32) s += Kb[n] * vb[n];
#pragma unroll
    for (int off = 16; off; off >>= 1) s += __shfl_xor(s, off, 32);
    if (lane == 0) u[row] = AW / (s + 1e-8f);
}

// ---------------- Sinkhorn: v = b / (u^T K + 1e-8), thread per column --------
// block 1024 = 4 m-groups x 256 columns; grid (8 col-chunks, 8 batches)
__global__ void sinkhorn_v_kernel(const float* __restrict__ Kbuf, const float* __restrict__ u,
                                  float* __restrict__ v) {
    __shared__ float red[1024];
    int b   = blockIdx.y;
    int col = blockIdx.x * 256 + (threadIdx.x & 255);
    int mg  = threadIdx.x >> 8;                       // 0..3
    const float* Kb = Kbuf + (size_t)b * NP * NP;
    const float* ub = u + b * NP;
    float s = 0.f;
    int m0 = mg * 512;
#pragma unroll 4
    for (int i = 0; i < 512; ++i) {
        int m = m0 + i;
        s += ub[m] * Kb[(size_t)m * NP + col];
    }
    red[threadIdx.x] = s;
    __syncthreads();
    if (mg == 0) {
        s = red[threadIdx.x] + red[threadIdx.x + 256] + red[threadIdx.x + 512] + red[threadIdx.x + 768];
        v[b * NP + col] = AW / (s + 1e-8f);
    }
}

// ---------------- final loss: online logsumexp over [l_pos, l_neg]/T ---------
// block 256 threads handles 64 rows of one batch; K row + K^T tile via LDS.
__global__ void loss_kernel(const float* __restrict__ Kbuf, const float* __restrict__ rq,
                            const float* __restrict__ rk, const float* __restrict__ u,
                            const float* __restrict__ v, float* __restrict__ out) {
    __shared__ float ldsA[64 * 65];   // K[m][n] tile
    __shared__ float ldsB[64 * 65];   // K[n][m] tile (transposed on store)
    __shared__ float sh_u[64], sh_rk[64];
    __shared__ float red_max[4 * 64], red_sum[4 * 64];

    int b  = blockIdx.y;
    int m0 = blockIdx.x * 64;
    const float* Kb = Kbuf + (size_t)b * NP * NP;
    int t = threadIdx.x;
    int m_local = t & 63;
    int grp     = t >> 6;             // 0..3, each covers 16 n per tile
    int m = m0 + m_local;

    float rq_m = rq[b * NP + m];
    float v_m  = v[b * NP + m];
    float runmax = -1e30f, runsum = 0.f;

    for (int n0 = 0; n0 < NP; n0 += 64) {
        __syncthreads();
#pragma unroll
        for (int i = 0; i < 16; ++i) {
            int idx = i * 256 + t;
            int r = idx >> 6, c = idx & 63;          // coalesced rows of K
            ldsA[r * 65 + c] = Kb[(size_t)(m0 + r) * NP + n0 + c];
            ldsB[c * 65 + r] = Kb[(size_t)(n0 + r) * NP + m0 + c];  // transpose into LDS
        }
        if (t < 64) { sh_u[t] = u[b * NP + n0 + t]; sh_rk[t] = rk[b * NP + n0 + t]; }
        __syncthreads();
#pragma unroll
        for (int j = 0; j < 16; ++j) {
            int n_local = grp * 16 + j;
            int n = n0 + n_local;
            float kmn = ldsA[m_local * 65 + n_local];   // K[m][n]
            float knm = ldsB[m_local * 65 + n_local];   // K[n][m]
            float S   = -__logf(kmn) * rq_m * sh_rk[n_local];          // q_m . k_n
            float fT  = sh_u[n_local] * knm * v_m * 2047.0f + 1e-8f;   // plan^T scaled
            float logit = (n == m) ? (-10.0f * INV_T) : (S * INV_T + __logf(fT));
            if (logit > runmax) {
                runsum = runsum * __expf(runmax - logit) + 1.0f;
                runmax = logit;
            } else {
                runsum += __expf(logit - runmax);
            }
        }
    }
    red_max[grp * 64 + m_local] = runmax;
    red_sum[grp * 64 + m_local] = runsum;
    __syncthreads();
    if (t < 64) {
        float kmm = Kb[(size_t)m * NP + m];
        float lp  = -__logf(kmm) * rq_m * rk[b * NP + m] * INV_T;   // l_pos / T
        float mx = lp;
#pragma unroll
        for (int g = 0; g < 4; ++g) mx = fmaxf(mx, red_max[g * 64 + t]);
        float s = __expf(lp - mx);
#pragma unroll
        for (int g = 0; g < 4; ++g) s += red_sum[g * 64 + t] * __expf(red_max[g * 64 + t] - mx);
        out[b * NP + m] = mx + __logf(s) - lp;       // logsumexp - l_pos/T
    }
}

extern "C" void kernel_launch(void* const* d_in, const int* in_sizes, int n_in,
                              void* d_out, int out_size, void* d_ws, size_t ws_size,
                              hipStream_t stream) {
    const float* fq = (const float*)d_in[0];
    const float* fk = (const float*)d_in[1];
    (void)in_sizes; (void)n_in; (void)out_size; (void)ws_size;   // shapes fixed: 8x2048x256
    float* out = (float*)d_out;

    // workspace layout: K (8*2048*2048 f32 = 134MB, L2-resident on MI455X), then norms/u/v
    float* Kbuf = (float*)d_ws;
    size_t KN = (size_t)NB * NP * NP;
    float* rq = Kbuf + KN;
    float* rk = rq + NROW;
    float* u  = rk + NROW;
    float* v  = u + NROW;

    norms_kernel<<<dim3(NROW / 8, 2), 256, 0, stream>>>(fq, fk, rq, rk);
    init_v_kernel<<<NROW / 256, 256, 0, stream>>>(v);
    gemm_expK_kernel<<<dim3(NP / 128, NP / 64, NB), 256, 0, stream>>>(fq, fk, rq, rk, Kbuf);
    for (int it = 0; it < 50; ++it) {
        sinkhorn_u_kernel<<<NROW / 8, 256, 0, stream>>>(Kbuf, v, u);
        sinkhorn_v_kernel<<<dim3(NP / 256, NB), 1024, 0, stream>>>(Kbuf, u, v);
    }
    loss_kernel<<<dim3(NP / 64, NB), 256, 0, stream>>>(Kbuf, rq, rk, u, v, out);
}